// RIRBox_MESH2IR_Hybrid_10960756540042
// MI455X (gfx1250) — compile-verified
//
#include <hip/hip_runtime.h>
#include <math.h>

// ---------------------------------------------------------------------------
// RIRBox/MESH2IR hybrid crossfade — memory-bound streaming kernel.
// HBM floor: ~195 MB / 23.3 TB/s ~= 8.4 us. No matrix structure -> no WMMA.
// CDNA5 path used: async global->LDS (ASYNCcnt) to stage the per-row-shifted
// mesh2ir stream; b128 traffic everywhere else; row-uniform math on SALU.
// ---------------------------------------------------------------------------

#ifndef __has_builtin
#define __has_builtin(x) 0
#endif

namespace {
constexpr int L        = 3968;  // shoebox row length == truncated mesh2ir length
constexpr int LM_IN    = 4800;  // raw mesh2ir row stride
constexpr int HALF     = 40;    // WIN // 2
constexpr int RAMPMAX  = 200;
constexpr int THREADS  = 992;   // 3968 / 4 float4-chunks; 992 = 31 * wave32
}

// Builtin signature (from hipcc diagnostic): param0 is a non-const AS1 pointer
// to "int __vector_size__(16)" (v4i). Mirror that exactly.
typedef int v4i __attribute__((vector_size(16)));
typedef __attribute__((address_space(1))) v4i glb_v4i;
typedef __attribute__((address_space(3))) v4i lds_v4i;

// 16B-aligned float4 vector for b128 load/store lowering.
typedef float fvec4 __attribute__((ext_vector_type(4), aligned(16)));

// ---- CDNA5 async global->LDS b128 (tracked by ASYNCcnt) -------------------
__device__ __forceinline__ void async_b128_to_lds(const float* gsrc, float* ldst) {
#if __has_builtin(__builtin_amdgcn_global_load_async_to_lds_b128)
  __builtin_amdgcn_global_load_async_to_lds_b128(
      (glb_v4i*)const_cast<float*>(gsrc),
      (lds_v4i*)ldst,
      /*offset=*/0, /*cpol=*/0);
#else
  unsigned lds_off = (unsigned)(unsigned long long)
      (__attribute__((address_space(3))) void*)ldst;
  unsigned long long ga = (unsigned long long)gsrc;
  asm volatile("global_load_async_to_lds_b128 %0, %1, off"
               :: "v"(lds_off), "v"(ga)
               : "memory");
#endif
}

__device__ __forceinline__ void wait_async0() {
#if __has_builtin(__builtin_amdgcn_s_wait_asynccnt)
  __builtin_amdgcn_s_wait_asynccnt(0);
#else
  asm volatile("s_wait_asynccnt 0" ::: "memory");
#endif
}

// ---------------------------------------------------------------------------

__global__ __launch_bounds__(THREADS)
void crossfade_kernel(const float* __restrict__ sb,
                      const float* __restrict__ m2,
                      const float* __restrict__ latent,
                      const int*   __restrict__ o_sb,
                      const int*   __restrict__ o_m,
                      float* __restrict__ out,
                      float* __restrict__ out_origins) {
  __shared__ float s_m2[L];  // 15,872 B: full truncated mesh2ir row

  const int b   = blockIdx.x;
  const int tid = threadIdx.x;
  const int j0  = tid * 4;

  // Kick off async stage of m2[b, 0:3968] into LDS: one aligned b128 per lane.
  // (row base b*4800*4 bytes is 16B aligned; tid*16 is 16B aligned)
  async_b128_to_lds(m2 + (size_t)b * LM_IN + j0, &s_m2[j0]);

  // Row-uniform scalars (blockIdx-indexed -> scalar loads / SALU / v_s_sqrt).
  const int   osb   = o_sb[b];
  const int   om    = o_m[b];
  const int   ss    = max(osb - HALF, 0);
  const int   ms    = max(om  - HALF, 0);
  const int   until = min(L - ss, L - ms);
  const float vol   = latent[b * 10 + 0] * latent[b * 10 + 1] * latent[b * 10 + 2];
  const int   mp    = (int)floorf(0.002f * sqrtf(vol) * 16000.0f);
  const bool  active = until > mp;
  const int   ramp  = min(max(until - mp, 0), RAMPMAX);
  const float denom = (float)max(ramp - 1, 1);
  const float inv_d = 1.0f / denom;   // one per-row divide instead of 4 per lane
  const int   delta = ms - ss;        // m_idx = j + delta (row-uniform shift)

  // Overlap: pull the aligned shoebox chunk while the async copy is in flight.
  const fvec4 sv = *(const fvec4*)(sb + (size_t)b * L + j0);

  wait_async0();      // this wave's ASYNCcnt == 0
  __syncthreads();    // all waves' staged data visible in LDS

  fvec4 r;
#pragma unroll
  for (int i = 0; i < 4; ++i) {
    const int   j  = j0 + i;
    const int   t  = j - ss;
    const int   k  = t - mp;
    const int   mi = min(max(j + delta, 0), L - 1);
    const float mv = s_m2[mi];                       // ds_load_b32
    const float w  = fminf(fmaxf((float)k * inv_d, 0.0f), 1.0f);
    const bool in_ramp = active && (k >= 0) && (k < ramp);
    const bool in_tail = active && (k >= ramp) && (t < until);
    float o = sv[i];
    o = in_tail ? mv : o;
    o = in_ramp ? (sv[i] * (1.0f - w) + mv * w) : o;
    r[i] = o;
  }
  *(fvec4*)(out + (size_t)b * L + j0) = r;           // global_store_b128

  // Second tuple output: origins = full(B, WIN//2) -> 40.0 in the flat buffer.
  if (tid == 0) out_origins[b] = 40.0f;
}

// ---------------------------------------------------------------------------

extern "C" void kernel_launch(void* const* d_in, const int* in_sizes, int n_in,
                              void* d_out, int out_size, void* d_ws, size_t ws_size,
                              hipStream_t stream) {
  const float* sb  = (const float*)d_in[0];  // (B, 3968)
  const float* m2  = (const float*)d_in[1];  // (B, 4800)
  const float* lat = (const float*)d_in[2];  // (B, 10)
  const int*   osb = (const int*)d_in[3];    // (B,)
  const int*   om  = (const int*)d_in[4];    // (B,)

  const int Bn = in_sizes[3];                // batch size from origin vector
  float* out         = (float*)d_out;
  float* out_origins = out + (size_t)Bn * L; // tuple output #2, flat after #1

  crossfade_kernel<<<dim3(Bn), dim3(THREADS), 0, stream>>>(
      sb, m2, lat, osb, om, out, out_origins);
}